// GraphSAGE_30889404793605
// MI455X (gfx1250) — compile-verified
//
#include <hip/hip_runtime.h>

typedef __attribute__((ext_vector_type(2))) float v2f;
typedef __attribute__((ext_vector_type(8))) float v8f;

#define NN 100000   // nodes
#define FEAT 128    // layer-1 feature width

// ---------------- zero fill ----------------
__global__ void zero_kernel(float* __restrict__ p, int n) {
  int i = blockIdx.x * blockDim.x + threadIdx.x;
  if (i < n) p[i] = 0.0f;
}

// ---------------- edge gather + scatter-add ----------------
// 32 lanes per edge; lane c moves features [4c, 4c+4) as a float4 and
// does 4 global atomic f32 adds (L2-resident destination array).
__global__ void scatter_kernel(const float* __restrict__ feat,     // [N,128]
                               const long long* __restrict__ src,  // [E]
                               const long long* __restrict__ dst,  // [E]
                               float* __restrict__ agg,            // [N,128]
                               float* __restrict__ cnt,            // [N]
                               int E, int addCnt) {
  long long t = (long long)blockIdx.x * blockDim.x + threadIdx.x;
  int e = (int)(t >> 5);
  int c = (int)(t & 31);
  if (e >= E) return;
  int s = (int)src[e];
  int d = (int)dst[e];
  const float4* fp = (const float4*)(feat + (size_t)s * FEAT);
  float4 v = fp[c];
  float* ap = agg + (size_t)d * FEAT + 4 * c;
  atomicAdd(ap + 0, v.x);
  atomicAdd(ap + 1, v.y);
  atomicAdd(ap + 2, v.z);
  atomicAdd(ap + 3, v.w);
  if (addCnt && c == 0) atomicAdd(cnt + d, 1.0f);
}

// ---------------- fused SAGE GEMM (fp32 WMMA) ----------------
// out[n][j] = sum_k (agg[n][k]/max(cnt[n],1)) * Wl[j][k]
//           + sum_k  Aself[n][k]              * Wr[j][k]
//           + bias[j]   (+ optional BN(running stats)+ReLU)
// One wave: 16 rows x (NJ*16) cols, K = 128 in steps of 4 via
// V_WMMA_F32_16X16X4_F32. A loaded once per k-step, reused for all NJ tiles.
template <int NJ, bool BN_RELU>
__global__ __launch_bounds__(256) void sage_gemm_kernel(
    const float* __restrict__ Aagg,   // [N,128] segment sums
    const float* __restrict__ cnt,    // [N] degrees
    const float* __restrict__ Aself,  // [N,128]
    const float* __restrict__ Wl,     // [NJ*16,128] row-major
    const float* __restrict__ Wr,     // [NJ*16,128] row-major
    const float* __restrict__ bias,   // [NJ*16]
    const float* __restrict__ gamma,
    const float* __restrict__ beta,
    const float* __restrict__ rmean,
    const float* __restrict__ rvar,
    float* __restrict__ out,          // [N, NJ*16]
    int nrows) {
  const int lane = threadIdx.x & 31;
  const int wave = threadIdx.x >> 5;
  const int rowBase = (blockIdx.x * 8 + wave) * 16;
  if (rowBase >= nrows) return;  // wave-uniform -> EXEC stays all-ones

  const int m  = lane & 15;          // A: row in tile / B,D: col in tile
  const int kh = (lane >> 4) * 2;    // K sub-offset for lane half (0 or 2)
  const int row = rowBase + m;

  const float scale = 1.0f / fmaxf(cnt[row], 1.0f);

  v8f acc[NJ] = {};
  const float* aggRow  = Aagg  + (size_t)row * 128 + kh;
  const float* selfRow = Aself + (size_t)row * 128 + kh;

  for (int k = 0; k < 128; k += 4) {
    v2f a1 = *(const v2f*)(aggRow + k);   // mean-path A tile
    a1 *= scale;                          // fold 1/deg into A
    v2f a2 = *(const v2f*)(selfRow + k);  // self-path A tile
#pragma unroll
    for (int j = 0; j < NJ; ++j) {
      const int col = j * 16 + m;
      v2f b1 = *(const v2f*)(Wl + (size_t)col * 128 + k + kh);
      v2f b2 = *(const v2f*)(Wr + (size_t)col * 128 + k + kh);
      acc[j] = __builtin_amdgcn_wmma_f32_16x16x4_f32(
          false, a1, false, b1, (short)0, acc[j], false, false);
      acc[j] = __builtin_amdgcn_wmma_f32_16x16x4_f32(
          false, a2, false, b2, (short)0, acc[j], false, false);
    }
  }

  // Epilogue: bias (+ BN + ReLU). Column index is per-lane constant in the
  // C/D layout; row = rowBase + r + 8*(lane>=16).
  const int rowOff = (lane >> 4) * 8;
  const int ldo = NJ * 16;
#pragma unroll
  for (int j = 0; j < NJ; ++j) {
    const int col = j * 16 + m;
    const float b = bias[col];
    float g = 0.f, be = 0.f, rm = 0.f, inv = 0.f;
    if (BN_RELU) {
      g = gamma[col];
      be = beta[col];
      rm = rmean[col];
      inv = rsqrtf(rvar[col] + 1e-5f);
    }
#pragma unroll
    for (int r = 0; r < 8; ++r) {
      float v = acc[j][r] + b;
      if (BN_RELU) {
        v = (v - rm) * inv * g + be;
        v = fmaxf(v, 0.0f);
      }
      out[(size_t)(rowBase + r + rowOff) * ldo + col] = v;
    }
  }
}

// ---------------- launcher ----------------
extern "C" void kernel_launch(void* const* d_in, const int* in_sizes, int n_in,
                              void* d_out, int out_size, void* d_ws,
                              size_t ws_size, hipStream_t stream) {
  const float* x        = (const float*)d_in[0];
  const long long* edge = (const long long*)d_in[1];  // int64 in reference
  const float* W1l   = (const float*)d_in[2];
  const float* b1l   = (const float*)d_in[3];
  const float* W1r   = (const float*)d_in[4];
  const float* gamma = (const float*)d_in[5];
  const float* beta  = (const float*)d_in[6];
  const float* rmean = (const float*)d_in[7];
  const float* rvar  = (const float*)d_in[8];
  const float* W2l   = (const float*)d_in[9];
  const float* b2l   = (const float*)d_in[10];
  const float* W2r   = (const float*)d_in[11];

  const int E = in_sizes[1] / 2;
  const long long* src = edge;
  const long long* dst = edge + E;

  float* ws  = (float*)d_ws;
  float* agg = ws;                          // [N,128]  (reused both layers)
  float* cnt = ws + (size_t)NN * 128;       // [N]
  float* h   = cnt + NN;                    // [N,128]
  float* out = (float*)d_out;               // [N,64]

  const int gemmBlocks = (NN / 16 + 7) / 8;                     // 782
  const int scatBlocks = (int)(((long long)E * 32 + 255) / 256);  // 200000

  // ---- layer 1 ----
  {
    int n = NN * 129;  // agg + cnt contiguous
    zero_kernel<<<(n + 255) / 256, 256, 0, stream>>>(agg, n);
  }
  scatter_kernel<<<scatBlocks, 256, 0, stream>>>(x, src, dst, agg, cnt, E, 1);
  sage_gemm_kernel<8, true><<<gemmBlocks, 256, 0, stream>>>(
      agg, cnt, x, W1l, W1r, b1l, gamma, beta, rmean, rvar, h, NN);

  // ---- layer 2 (degrees reused; re-zero agg only) ----
  {
    int n = NN * 128;
    zero_kernel<<<(n + 255) / 256, 256, 0, stream>>>(agg, n);
  }
  scatter_kernel<<<scatBlocks, 256, 0, stream>>>(h, src, dst, agg, cnt, E, 0);
  sage_gemm_kernel<4, false><<<gemmBlocks, 256, 0, stream>>>(
      agg, cnt, h, W2l, W2r, b2l, nullptr, nullptr, nullptr, nullptr, out, NN);
}